// Basic_Net_3221225472394
// MI455X (gfx1250) — compile-verified
//
#include <hip/hip_runtime.h>
#include <hip/hip_bf16.h>

// ---------------------------------------------------------------------------
// GCN (3x GCNConv + MLP head + weighted BCE) for MI455X (gfx1250, wave32).
// Dense GEMMs use V_WMMA_F32_16X16X4_F32 (fp32 in / fp32 accumulate).
// Edge aggregation uses wave-per-edge coalesced gathers + f32 global atomics
// (entire 25.6MB feature matrix lives in the 192MB L2).
// ---------------------------------------------------------------------------

typedef float v2f __attribute__((ext_vector_type(2)));
typedef float v8f __attribute__((ext_vector_type(8)));

#define HID 64

// ---- degree / normalization ------------------------------------------------

__global__ void init_deg_kernel(float* __restrict__ deg, int N) {
  int v = blockIdx.x * blockDim.x + threadIdx.x;
  if (v < N) deg[v] = 1.0f;  // self loop
}

__global__ void deg_accum_kernel(const long long* __restrict__ col,
                                 float* __restrict__ deg, int E) {
  int stride = gridDim.x * blockDim.x;
  for (int e = blockIdx.x * blockDim.x + threadIdx.x; e < E; e += stride)
    atomicAdd(&deg[(int)col[e]], 1.0f);
}

__global__ void dinv_kernel(float* __restrict__ deg, int N) {
  int v = blockIdx.x * blockDim.x + threadIdx.x;
  if (v < N) deg[v] = rsqrtf(deg[v]);  // overwrite deg with D^{-1/2}
}

// ---- dense GEMM: C[N x 64] = A[N x K] * B[K x 64], fp32 WMMA ---------------
// One wave computes one 16x16 tile; 4 waves/block cover the 64 output cols.

__global__ void gemm_wmma_kernel(const float* __restrict__ A,
                                 const float* __restrict__ B,
                                 float* __restrict__ C, int N, int K) {
  const int lane = threadIdx.x & 31;
  const int wave = threadIdx.x >> 5;          // 0..3 -> 16-col slice
  const int rowBlock = blockIdx.x << 4;       // 16 rows per block
  const int m = lane & 15;
  const int kOff = (lane >> 4) << 1;          // lanes 0-15: K+0/1, 16-31: K+2/3
  const int n = (lane & 15) + (wave << 4);

  int mRow = rowBlock + m;
  if (mRow >= N) mRow = N - 1;                // clamp loads; EXEC stays all-1s
  const float* arow = A + (size_t)mRow * K + kOff;
  const float* bcol = B + n + kOff * HID;

  v8f acc = {};
  for (int k0 = 0; k0 < K; k0 += 4) {
    v2f a, b;
    a.x = arow[k0];
    a.y = arow[k0 + 1];
    b.x = bcol[k0 * HID];
    b.y = bcol[(k0 + 1) * HID];
    acc = __builtin_amdgcn_wmma_f32_16x16x4_f32(
        /*neg_a=*/false, a, /*neg_b=*/false, b,
        /*c_mod=*/(short)0, acc, /*reuse_a=*/false, /*reuse_b=*/false);
  }

  // C/D layout: VGPR r -> row r (lanes 0-15) / row r+8 (lanes 16-31)
  const int mBase = rowBlock + ((lane >> 4) << 3);
#pragma unroll
  for (int r = 0; r < 8; ++r) {
    int row = mBase + r;
    if (row < N) C[(size_t)row * HID + n] = acc[r];
  }
}

// ---- acc[v][j] = b[j] + dinv[v]^2 * HW[v][j]  (self-loop term + bias) ------

__global__ void init_acc_kernel(const float* __restrict__ HW,
                                const float* __restrict__ dinv,
                                const float* __restrict__ b,
                                float* __restrict__ acc, int N) {
  int total = N * HID;
  int stride = gridDim.x * blockDim.x;
  for (int i = blockIdx.x * blockDim.x + threadIdx.x; i < total; i += stride) {
    int v = i >> 6;
    int j = i & 63;
    float d = dinv[v];
    acc[i] = b[j] + d * d * HW[i];
  }
}

// ---- edge scatter: wave-per-edge, lane j handles features j and j+32 -------

__global__ void scatter_kernel(const long long* __restrict__ rows,
                               const long long* __restrict__ cols,
                               const float* __restrict__ dinv,
                               const float* __restrict__ HW,
                               float* __restrict__ acc, int E) {
  int lane = threadIdx.x & 31;
  int gw = (blockIdx.x * blockDim.x + threadIdx.x) >> 5;
  int nW = (gridDim.x * blockDim.x) >> 5;
  for (int e = gw; e < E; e += nW) {
    int r = (int)rows[e];
    int c = (int)cols[e];
    float nrm = dinv[r] * dinv[c];
    const float* src = HW + (size_t)r * HID;
    float* dst = acc + (size_t)c * HID;
    atomicAdd(&dst[lane],      nrm * src[lane]);
    atomicAdd(&dst[lane + 32], nrm * src[lane + 32]);
  }
}

__global__ void relu_kernel(const float* __restrict__ acc,
                            float* __restrict__ h, int total) {
  int stride = gridDim.x * blockDim.x;
  for (int i = blockIdx.x * blockDim.x + threadIdx.x; i < total; i += stride)
    h[i] = fmaxf(acc[i], 0.0f);
}

// ---- loss bookkeeping ------------------------------------------------------

__global__ void zero_scalars_kernel(float* __restrict__ s) {
  if (blockIdx.x == 0 && threadIdx.x == 0) { s[0] = 0.0f; s[1] = 0.0f; }
}

__global__ void label_sum_kernel(const int* __restrict__ labels,
                                 float* __restrict__ scal, int N) {
  int stride = gridDim.x * blockDim.x;
  float local = 0.0f;
  for (int v = blockIdx.x * blockDim.x + threadIdx.x; v < N; v += stride)
    local += (float)labels[v];
  atomicAdd(&scal[0], local);
}

// ---- MLP head (64 -> 8 -> 1), sigmoid, weighted BCE ------------------------

__global__ void head_kernel(const float* __restrict__ h,
                            const int* __restrict__ labels,
                            const float* __restrict__ lW1,
                            const float* __restrict__ lb1,
                            const float* __restrict__ lW2,
                            const float* __restrict__ lb2,
                            float* __restrict__ scal,  // [0]=labelSum [1]=lossAcc
                            float* __restrict__ out,   // out[0]=loss, out[1+v]=p
                            int N) {
  int v = blockIdx.x * blockDim.x + threadIdx.x;
  if (v >= N) return;
  const float* hv = h + (size_t)v * HID;

  float z[8];
#pragma unroll
  for (int j = 0; j < 8; ++j) z[j] = lb1[j];
  for (int k = 0; k < HID; ++k) {
    float hk = hv[k];
#pragma unroll
    for (int j = 0; j < 8; ++j) z[j] += hk * lW1[k * 8 + j];
  }
  float s = lb2[0];
#pragma unroll
  for (int j = 0; j < 8; ++j) s += fmaxf(z[j], 0.0f) * lW2[j];

  float p = 1.0f / (1.0f + expf(-s));
  out[1 + v] = p;

  float y = (float)labels[v];
  float pm = scal[0] / (float)N;
  float w = y * (1.0f - pm) + (1.0f - y) * pm;
  const float eps = 1e-7f;
  float pc = fminf(fmaxf(p, eps), 1.0f - eps);
  float bce = -(y * logf(pc) + (1.0f - y) * logf(1.0f - pc));
  atomicAdd(&scal[1], w * bce);
}

__global__ void finalize_kernel(const float* __restrict__ scal,
                                float* __restrict__ out, int N) {
  if (blockIdx.x == 0 && threadIdx.x == 0)
    out[0] = scal[1] / (float)N;
}

// ---------------------------------------------------------------------------

extern "C" void kernel_launch(void* const* d_in, const int* in_sizes, int n_in,
                              void* d_out, int out_size, void* d_ws, size_t ws_size,
                              hipStream_t stream) {
  const float*     x      = (const float*)d_in[0];
  const long long* ei     = (const long long*)d_in[1];  // jnp.int64, shape (2,E)
  const int*       labels = (const int*)d_in[2];
  const float*     W1 = (const float*)d_in[3];
  const float*     b1 = (const float*)d_in[4];
  const float*     W2 = (const float*)d_in[5];
  const float*     b2 = (const float*)d_in[6];
  const float*     W3 = (const float*)d_in[7];
  const float*     b3 = (const float*)d_in[8];
  const float*     lW1 = (const float*)d_in[9];
  const float*     lb1 = (const float*)d_in[10];
  const float*     lW2 = (const float*)d_in[11];
  const float*     lb2 = (const float*)d_in[12];

  const int N  = in_sizes[2];          // labels is [N,1]
  const int E  = in_sizes[1] / 2;      // edge_index is [2,E]
  const int K0 = in_sizes[0] / N;      // 128 input features

  const long long* rows = ei;          // sources
  const long long* cols = ei + E;      // targets

  float* ws   = (float*)d_ws;
  float* dinv = ws;                           // N   (deg, then D^{-1/2})
  float* HW   = dinv + N;                     // N*64
  float* acc  = HW + (size_t)N * HID;         // N*64
  float* h    = acc + (size_t)N * HID;        // N*64
  float* scal = h + (size_t)N * HID;          // 2 floats
  float* out  = (float*)d_out;

  const int T = 256;
  const int nBlkN  = (N + T - 1) / T;
  const int nBlkNH = 2048;

  // normalization
  init_deg_kernel<<<nBlkN, T, 0, stream>>>(dinv, N);
  deg_accum_kernel<<<2048, T, 0, stream>>>(cols, dinv, E);
  dinv_kernel<<<nBlkN, T, 0, stream>>>(dinv, N);

  const float* Ws[3] = {W1, W2, W3};
  const float* bs[3] = {b1, b2, b3};
  const float* in = x;
  int K = K0;
  for (int l = 0; l < 3; ++l) {
    gemm_wmma_kernel<<<(N + 15) / 16, 128, 0, stream>>>(in, Ws[l], HW, N, K);
    init_acc_kernel<<<nBlkNH, T, 0, stream>>>(HW, dinv, bs[l], acc, N);
    scatter_kernel<<<8192, T, 0, stream>>>(rows, cols, dinv, HW, acc, E);
    relu_kernel<<<nBlkNH, T, 0, stream>>>(acc, h, N * HID);
    in = h;
    K = HID;
  }

  zero_scalars_kernel<<<1, 1, 0, stream>>>(scal);
  label_sum_kernel<<<nBlkN, T, 0, stream>>>(labels, scal, N);
  head_kernel<<<nBlkN, T, 0, stream>>>(h, labels, lW1, lb1, lW2, lb2, scal, out, N);
  finalize_kernel<<<1, 1, 0, stream>>>(scal, out, N);
}